// GCN2_30562987278375
// MI455X (gfx1250) — compile-verified
//
#include <hip/hip_runtime.h>
#include <math.h>

typedef __attribute__((ext_vector_type(2))) float v2f;
typedef __attribute__((ext_vector_type(8))) float v8f;

#define HIDC 128   // hidden width (matches reference HID)

// ---------------------------------------------------------------- utilities
__global__ void fill_kernel(float* __restrict__ p, float v, long long n) {
  long long i = (long long)blockIdx.x * blockDim.x + threadIdx.x;
  if (i < n) p[i] = v;
}

__global__ void deg_acc_kernel(const int* __restrict__ dst, float* __restrict__ deg, int E) {
  int e = blockIdx.x * blockDim.x + threadIdx.x;
  if (e < E) atomicAdd(&deg[dst[e]], 1.0f);
}

__global__ void rsqrt_kernel(float* __restrict__ deg, int n) {
  int i = blockIdx.x * blockDim.x + threadIdx.x;
  if (i < n) {
    float d = deg[i];
    deg[i] = (d > 0.0f) ? rsqrtf(d) : 0.0f;
  }
}

__global__ void norm_kernel(const int* __restrict__ src, const int* __restrict__ dst,
                            const float* __restrict__ dinv, float* __restrict__ nrm, int E) {
  int e = blockIdx.x * blockDim.x + threadIdx.x;
  if (e < E) nrm[e] = dinv[src[e]] * dinv[dst[e]];
}

// One wave (32 lanes) per edge; each lane moves a float4 of channels.
// agg[dst] += norm[e] * h[src]   (self-loop term is folded into z_kernel)
__global__ void scatter_kernel(const int* __restrict__ src, const int* __restrict__ dst,
                               const float* __restrict__ nrm, const float* __restrict__ h,
                               float* __restrict__ agg, int E) {
  long long t = (long long)blockIdx.x * blockDim.x + threadIdx.x;
  int e = (int)(t >> 5);
  if (e >= E) return;
  int c = ((int)t & 31) << 2;
  int s = src[e], d = dst[e];
  float w = nrm[e];
  const float4 hv = *(const float4*)(h + (size_t)s * HIDC + c);
  float* ap = agg + (size_t)d * HIDC + c;
  atomicAdd(ap + 0, w * hv.x);
  atomicAdd(ap + 1, w * hv.y);
  atomicAdd(ap + 2, w * hv.z);
  atomicAdd(ap + 3, w * hv.w);
}

// z = (1-alpha) * (agg + dinv^2 * h) + alpha * h0   (written in place over agg)
__global__ void z_kernel(float* __restrict__ agg, const float* __restrict__ h,
                         const float* __restrict__ h0, const float* __restrict__ dinv,
                         float alpha, long long n) {
  long long i = (long long)blockIdx.x * blockDim.x + threadIdx.x;
  if (i >= n) return;
  int node = (int)(i >> 7);  // /HIDC
  float di = dinv[node];
  float z = (1.0f - alpha) * (agg[i] + di * di * h[i]) + alpha * h0[i];
  agg[i] = z;
}

// ---------------------------------------------------------------- WMMA GEMM
// Out[M x Ncols] = post( ew * Z + mm * (Z @ W) + bias )
//   Z: [M x K] row-major, W: [K x Ncols] row-major.
// Block = 128 threads = 4 waves; each wave owns a 16x16 output tile.
// The 16-column slice of W for this block is staged in LDS (K*16 floats).
// Uses V_WMMA_F32_16X16X4_F32 (fp32-exact, CDNA5 matrix pipe).
__global__ void wmma_gemm_kernel(const float* __restrict__ Z, const float* __restrict__ W,
                                 const float* __restrict__ bias, float* __restrict__ Out,
                                 int M, int K, int Ncols,
                                 float ew, float mm, int do_relu) {
  extern __shared__ float ldsW[];  // [K][16]
  const int n0 = blockIdx.y * 16;

  for (int i = threadIdx.x; i < K * 16; i += blockDim.x) {
    int kk = i >> 4;
    int nn = i & 15;
    ldsW[i] = W[(size_t)kk * Ncols + n0 + nn];
  }
  __syncthreads();

  const int lane = threadIdx.x & 31;
  const int wave = threadIdx.x >> 5;
  const int m0 = (blockIdx.x * 4 + wave) * 16;
  if (m0 >= M) return;  // wave-uniform: EXEC stays all-ones for live waves

  // A 16x4 f32 layout: lanes 0-15 -> M=lane, K={0,1}; lanes 16-31 -> M=lane-16, K={2,3}
  const int lrow  = lane & 15;
  const int khalf = (lane >> 4) << 1;
  const float* zr = Z + (size_t)(m0 + lrow) * K + khalf;
  const float* wr = ldsW + khalf * 16 + lrow;

  v8f acc = {};
  for (int k = 0; k < K; k += 4) {
    v2f a, b;
    a.x = zr[k];
    a.y = zr[k + 1];
    b.x = wr[k * 16];
    b.y = wr[k * 16 + 16];
    acc = __builtin_amdgcn_wmma_f32_16x16x4_f32(
        /*neg_a=*/false, a, /*neg_b=*/false, b,
        /*c_mod=*/(short)0, acc, /*reuse_a=*/false, /*reuse_b=*/false);
  }

  // C/D layout: VGPR r holds (M = m0 + r + 8*(lane>=16), N = n0 + (lane&15))
  const int mbase = m0 + ((lane >> 4) << 3);
  const int nc = n0 + lrow;
  const float bv = bias ? bias[nc] : 0.0f;
#pragma unroll
  for (int r = 0; r < 8; ++r) {
    const int m = mbase + r;
    float v = mm * acc[r] + bv;
    if (ew != 0.0f) v += ew * Z[(size_t)m * K + nc];  // only when K == Ncols (conv layers)
    if (do_relu) v = fmaxf(v, 0.0f);
    Out[(size_t)m * Ncols + nc] = v;
  }
}

// ---------------------------------------------------------------- launcher
extern "C" void kernel_launch(void* const* d_in, const int* in_sizes, int n_in,
                              void* d_out, int out_size, void* d_ws, size_t ws_size,
                              hipStream_t stream) {
  const float* x      = (const float*)d_in[0];
  const int*   ei     = (const int*)  d_in[1];
  const float* W_in   = (const float*)d_in[2];
  const float* b_in   = (const float*)d_in[3];
  const float* W_conv = (const float*)d_in[4];
  const float* W_out  = (const float*)d_in[5];
  const float* b_out  = (const float*)d_in[6];

  const int HID   = in_sizes[3];                 // 128
  const int OUTC  = in_sizes[6];                 // 64
  const int INC   = in_sizes[2] / HID;           // 500
  const int Nn    = in_sizes[0] / INC;           // 100000
  const int E     = in_sizes[1] / 2;             // 1600000
  const int L     = in_sizes[4] / (HID * HID);   // 8
  const int* srcp = ei;
  const int* dstp = ei + E;

  float* ws   = (float*)d_ws;
  float* h    = ws;                              // N*HID
  float* h0   = h   + (size_t)Nn * HID;          // N*HID
  float* agg  = h0  + (size_t)Nn * HID;          // N*HID  (reused as z)
  float* dinv = agg + (size_t)Nn * HID;          // N      (deg -> dinv in place)
  float* nrm  = dinv + Nn;                       // E

  const int tpb = 256;

  // gcn_norm: deg (self-loop -> init 1.0), rsqrt, per-edge norm
  fill_kernel<<<(Nn + tpb - 1) / tpb, tpb, 0, stream>>>(dinv, 1.0f, Nn);
  deg_acc_kernel<<<(E + tpb - 1) / tpb, tpb, 0, stream>>>(dstp, dinv, E);
  rsqrt_kernel<<<(Nn + tpb - 1) / tpb, tpb, 0, stream>>>(dinv, Nn);
  norm_kernel<<<(E + tpb - 1) / tpb, tpb, 0, stream>>>(srcp, dstp, dinv, nrm, E);

  const int tiles = (Nn + 15) / 16;         // 6250 (16 | 100000 -> no partial tiles)
  const int rowg  = (tiles + 3) / 4;        // 4 tiles per block (4 waves)

  // h0 = relu(x @ W_in + b_in)
  dim3 gin(rowg, HID / 16);
  wmma_gemm_kernel<<<gin, 128, (size_t)INC * 16 * sizeof(float), stream>>>(
      x, W_in, b_in, h0, Nn, INC, HID, 0.0f, 1.0f, 1);
  hipMemcpyAsync(h, h0, (size_t)Nn * HID * sizeof(float),
                 hipMemcpyDeviceToDevice, stream);

  const long long nh = (long long)Nn * HID;
  const long long st = (long long)E * 32;
  for (int l = 0; l < L; ++l) {
    fill_kernel<<<(int)((nh + tpb - 1) / tpb), tpb, 0, stream>>>(agg, 0.0f, nh);
    scatter_kernel<<<(int)((st + tpb - 1) / tpb), tpb, 0, stream>>>(
        srcp, dstp, nrm, h, agg, E);
    z_kernel<<<(int)((nh + tpb - 1) / tpb), tpb, 0, stream>>>(
        agg, h, h0, dinv, 0.1f, nh);
    const float beta = logf(0.5f / (float)(l + 1) + 1.0f);
    wmma_gemm_kernel<<<gin, 128, (size_t)HID * 16 * sizeof(float), stream>>>(
        agg, W_conv + (size_t)l * HID * HID, nullptr, h,
        Nn, HID, HID, 1.0f - beta, beta, 1);
  }

  // out = h @ W_out + b_out
  dim3 gout(rowg, OUTC / 16);
  wmma_gemm_kernel<<<gout, 128, (size_t)HID * 16 * sizeof(float), stream>>>(
      h, W_out, b_out, (float*)d_out, Nn, HID, OUTC, 0.0f, 1.0f, 0);
}